// NetVLAD_10307921510983
// MI455X (gfx1250) — compile-verified
//
#include <hip/hip_runtime.h>
#include <math.h>

// Problem constants (fixed by the reference): N=64, C=512, H*W=P=1024, K=64
#define N_IMG 64
#define C_DIM 512
#define P_DIM 1024
#define K_CL  64

typedef __attribute__((ext_vector_type(2))) float v2f;
typedef __attribute__((ext_vector_type(4))) float v4f;
typedef __attribute__((ext_vector_type(8))) float v8f;

// ---------------- workspace layout (float offsets) ----------------
// a_t      : [N][P][K]   transposed softmax assignments (coalesced for GEMM2)
// asum     : [N][16][K]  per-pixel-block partial sums of a over p
// vlad_pre : [N][K][C]   post-residual, pre-normalization VLAD
// sumsq    : [N][K][8]   per-c-block partial sum of squares
#define WS_AT   ((size_t)0)
#define WS_ASUM (WS_AT   + (size_t)N_IMG * P_DIM * K_CL)
#define WS_VLAD (WS_ASUM + (size_t)N_IMG * 16 * K_CL)
#define WS_SQ   (WS_VLAD + (size_t)N_IMG * K_CL * C_DIM)

__device__ __forceinline__ v8f wmma_f32(v2f a, v2f b, v8f c) {
  // D = A(16x4 f32) * B(4x16 f32) + C(16x16 f32); exact f32 FMA math
  return __builtin_amdgcn_wmma_f32_16x16x4_f32(
      /*neg_a=*/false, a, /*neg_b=*/false, b,
      /*c_mod=*/(short)0, c, /*reuse_a=*/false, /*reuse_b=*/false);
}

// =====================================================================
// Kernel 1: logits = W @ x[n] + b  (per 64-pixel block), softmax over K,
//           store a transposed as a_t[n][p][k], and per-block asum[k].
// grid = (16 pixel-blocks, 64 images), block = 256 (8 waves)
// =====================================================================
__global__ __launch_bounds__(256)
void k1_logits_softmax(const float* __restrict__ x,
                       const float* __restrict__ w,
                       const float* __restrict__ bias,
                       float* __restrict__ ws) {
  __shared__ float lds[64 * 65];          // [p_local][k], stride 65 (bank-friendly)

  const int n   = blockIdx.y;
  const int pb  = blockIdx.x;             // pixel block 0..15
  const int tid = threadIdx.x;
  const int wv  = tid >> 5;               // wave 0..7 (wave32)
  const int lane = tid & 31;
  const int lr  = lane & 15;              // lane 0..15 within half
  const int lh  = lane >> 4;              // half 0/1 (carries K=2,3 per ISA A layout)

  const int m0  = (wv & 3) * 16;          // cluster-row tile base
  const int npair = wv >> 2;              // which pair of pixel tiles
  const int p0a = pb * 64 + (2 * npair) * 16;
  const int p0b = p0a + 16;

  const float* xn = x + (size_t)n * C_DIM * P_DIM;

  v8f acc0 = {}; v8f acc1 = {};
  const int arow = (m0 + lr) * C_DIM + 2 * lh;  // conv_w row, K-offset per half
  #pragma unroll 4
  for (int c = 0; c < C_DIM; c += 4) {
    // A fragment: conv_w[m0+lr][c+2*lh], [c+2*lh+1]  -> one b64 load
    v2f a = *(const v2f*)(w + arow + c);
    // B fragments: x[c+2*lh][p], x[c+2*lh+1][p] (rows are P floats apart)
    const float* xc = xn + (size_t)(c + 2 * lh) * P_DIM;
    v2f b0, b1;
    b0.x = xc[p0a + lr]; b0.y = xc[P_DIM + p0a + lr];
    b1.x = xc[p0b + lr]; b1.y = xc[P_DIM + p0b + lr];
    acc0 = wmma_f32(a, b0, acc0);
    acc1 = wmma_f32(a, b1, acc1);
  }

  // Add bias, stash logits to LDS as [p_local][k]
  const int pl0 = (2 * npair) * 16 + lr;
  #pragma unroll
  for (int v = 0; v < 8; ++v) {
    const int k = m0 + v + 8 * lh;        // C/D layout: M = v + 8*(lane/16)
    const float bv = bias[k];
    lds[(pl0)      * 65 + k] = acc0[v] + bv;
    lds[(pl0 + 16) * 65 + k] = acc1[v] + bv;
  }
  __syncthreads();

  // Softmax over K=64 per pixel (threads 0..63, one pixel each)
  if (tid < 64) {
    float* row = lds + tid * 65;
    float m = row[0];
    #pragma unroll 8
    for (int k = 1; k < 64; ++k) m = fmaxf(m, row[k]);
    float s = 0.f;
    #pragma unroll 8
    for (int k = 0; k < 64; ++k) { float e = __expf(row[k] - m); row[k] = e; s += e; }
    const float inv = 1.0f / s;
    #pragma unroll 8
    for (int k = 0; k < 64; ++k) row[k] *= inv;
  }
  __syncthreads();

  // Per-block asum[k] = sum_p a[p][k]  (threads 0..63, one k each)
  if (tid < 64) {
    float s = 0.f;
    #pragma unroll 8
    for (int p = 0; p < 64; ++p) s += lds[p * 65 + tid];
    ws[WS_ASUM + ((size_t)n * 16 + pb) * K_CL + tid] = s;
  }

  // Coalesced store of a_t[n][p][k]: 4 threads per pixel row, b128 stores
  {
    const int p = tid >> 2, q = tid & 3;
    float* dst = ws + WS_AT + ((size_t)n * P_DIM + pb * 64 + p) * K_CL + q * 16;
    const float* src = lds + p * 65 + q * 16;
    #pragma unroll
    for (int i = 0; i < 4; ++i) {
      v4f vv; vv.x = src[4*i]; vv.y = src[4*i+1]; vv.z = src[4*i+2]; vv.w = src[4*i+3];
      *(v4f*)(dst + 4 * i) = vv;
    }
  }
}

// =====================================================================
// Kernel 2: vlad_pre[n][k][c] = sum_p a[k][p]*x[c][p] - asum[k]*cent[k][c]
//           plus deterministic per-(k, c-block) sum-of-squares partials.
// grid = (8 c-blocks, 64 images), block = 256 (8 waves)
// =====================================================================
__global__ __launch_bounds__(256)
void k2_vlad(const float* __restrict__ x,
             const float* __restrict__ cent,
             float* __restrict__ ws) {
  __shared__ float s_sq[64 * 65];         // [k][c_local] squares, stride 65
  __shared__ float s_asum[64];

  const int n   = blockIdx.y;
  const int cb  = blockIdx.x;             // channel block 0..7 (64 channels each)
  const int tid = threadIdx.x;
  const int wv  = tid >> 5;
  const int lane = tid & 31;
  const int lr  = lane & 15;
  const int lh  = lane >> 4;

  const int m0 = (wv & 3) * 16;           // cluster-row tile
  const int cpair = wv >> 2;
  const int c0a = cb * 64 + (2 * cpair) * 16;
  const int c0b = c0a + 16;

  // Reduce the 16 per-block asum partials (deterministic, no atomics)
  if (tid < 64) {
    float s = 0.f;
    #pragma unroll 4
    for (int blk = 0; blk < 16; ++blk)
      s += ws[WS_ASUM + ((size_t)n * 16 + blk) * K_CL + tid];
    s_asum[tid] = s;
  }
  __syncthreads();

  const float* xn = x + (size_t)n * C_DIM * P_DIM;
  const float* at = ws + WS_AT + (size_t)n * P_DIM * K_CL;

  v8f acc0 = {}; v8f acc1 = {};
  #pragma unroll 4
  for (int p = 0; p < P_DIM; p += 4) {
    // A fragment: a[k=m0+lr][p+2*lh], [p+2*lh+1] via a_t -> coalesced b32 x2
    v2f a;
    a.x = at[(size_t)(p + 2 * lh)     * K_CL + m0 + lr];
    a.y = at[(size_t)(p + 2 * lh + 1) * K_CL + m0 + lr];
    // B fragments: x[c][p+2*lh..+1] -> per-lane b64 (consecutive p)
    v2f b0 = *(const v2f*)(xn + (size_t)(c0a + lr) * P_DIM + p + 2 * lh);
    v2f b1 = *(const v2f*)(xn + (size_t)(c0b + lr) * P_DIM + p + 2 * lh);
    acc0 = wmma_f32(a, b0, acc0);
    acc1 = wmma_f32(a, b1, acc1);
  }

  // Residual subtraction, store raw vlad, stash squares for norm
  #pragma unroll
  for (int v = 0; v < 8; ++v) {
    const int k  = m0 + v + 8 * lh;
    const float as = s_asum[k];
    const int ca = c0a + lr, cbx = c0b + lr;
    const float va = acc0[v] - as * cent[k * C_DIM + ca];
    const float vb = acc1[v] - as * cent[k * C_DIM + cbx];
    ws[WS_VLAD + ((size_t)n * K_CL + k) * C_DIM + ca]  = va;
    ws[WS_VLAD + ((size_t)n * K_CL + k) * C_DIM + cbx] = vb;
    s_sq[k * 65 + (ca  - cb * 64)] = va * va;
    s_sq[k * 65 + (cbx - cb * 64)] = vb * vb;
  }
  __syncthreads();

  if (tid < 64) {
    float s = 0.f;
    const float* row = s_sq + tid * 65;
    #pragma unroll 8
    for (int i = 0; i < 64; ++i) s += row[i];
    ws[WS_SQ + ((size_t)n * K_CL + tid) * 8 + cb] = s;
  }
}

// =====================================================================
// Kernel 3: intra-normalize per (n,k), global-normalize per n, write out.
// grid = 64 images, block = 256
// =====================================================================
__global__ __launch_bounds__(256)
void k3_normalize(const float* __restrict__ ws, float* __restrict__ out) {
  __shared__ float s_rinv[64];
  __shared__ float s_contrib[64];
  __shared__ float s_Rinv;

  const int n = blockIdx.x, tid = threadIdx.x;
  if (tid < 64) {
    float r2 = 0.f;
    #pragma unroll
    for (int cb = 0; cb < 8; ++cb)
      r2 += ws[WS_SQ + ((size_t)n * K_CL + tid) * 8 + cb];
    const float r    = sqrtf(r2);
    const float rinv = 1.0f / fmaxf(r, 1e-12f);
    s_rinv[tid]    = rinv;
    s_contrib[tid] = r2 * rinv * rinv;    // ||row/max(r,eps)||^2
  }
  __syncthreads();
  if (tid == 0) {
    float R2 = 0.f;
    #pragma unroll 8
    for (int k = 0; k < 64; ++k) R2 += s_contrib[k];
    s_Rinv = 1.0f / fmaxf(sqrtf(R2), 1e-12f);
  }
  __syncthreads();

  const float Rinv = s_Rinv;
  const float* src = ws + WS_VLAD + (size_t)n * K_CL * C_DIM;
  float* dst = out + (size_t)n * K_CL * C_DIM;
  for (int i = tid * 4; i < K_CL * C_DIM; i += 256 * 4) {
    v4f vv = *(const v4f*)(src + i);
    const float f = s_rinv[i >> 9] * Rinv;   // 512 floats per k-row, 4-aligned
    vv.x *= f; vv.y *= f; vv.z *= f; vv.w *= f;
    *(v4f*)(dst + i) = vv;
  }
}

// =====================================================================
extern "C" void kernel_launch(void* const* d_in, const int* in_sizes, int n_in,
                              void* d_out, int out_size, void* d_ws, size_t ws_size,
                              hipStream_t stream) {
  const float* x    = (const float*)d_in[0];  // [64,512,32,32]
  const float* w    = (const float*)d_in[1];  // [64,512]
  const float* b    = (const float*)d_in[2];  // [64]
  const float* cent = (const float*)d_in[3];  // [64,512]
  float* ws  = (float*)d_ws;
  float* out = (float*)d_out;                 // [64, 64*512]

  k1_logits_softmax<<<dim3(16, N_IMG), 256, 0, stream>>>(x, w, b, ws);
  k2_vlad<<<dim3(8, N_IMG), 256, 0, stream>>>(x, cent, ws);
  k3_normalize<<<dim3(N_IMG), 256, 0, stream>>>(ws, out);
}